// TransitionDown_85461259256090
// MI455X (gfx1250) — compile-verified
//
#include <hip/hip_runtime.h>
#include <hip/hip_bf16.h>
#include <math.h>

// ---------------------------------------------------------------------------
// TransitionDown for MI455X (gfx1250, wave32).
//   feats  [8,8192,64] f32 | points [8,8192,3] f32 | W [67,128] | b_fc,gamma,beta [128]
//   out = concat( x[8,2048,128], p2_points[8,2048,3] ) flat f32
// Pipeline: FPS (serial scan) -> kNN (top-16) -> WMMA f32 GEMM (K=67 pad 68)
//           -> global BN stats -> BN+GELU+max-over-k.
// GEMM uses V_WMMA_F32_16X16X4_F32; W kept TRANSPOSED in LDS so every A/B
// fragment is one aligned ds_load_b64 (bank-conflict-free), and each wave
// processes 4 tiles per block to amortize W staging.
// ---------------------------------------------------------------------------

typedef __attribute__((ext_vector_type(2))) float v2f;
typedef __attribute__((ext_vector_type(8))) float v8f;

#define B     8
#define N     8192
#define NP    2048
#define CIN   64
#define COUT  128
#define KNNK  16
#define KDIM  67
#define KPAD  68
#define NROW  (B*NP*KNNK)     // 262144 GEMM rows
#define NTILE (NROW/16)       // 16384 16-row tiles
#define TPW   4               // tiles per wave in gemm kernel

// workspace byte offsets
#define OFF_P2IDX 0
#define OFF_KNN   (OFF_P2IDX + B*NP*4)        // 65536
#define OFF_SUMS  (OFF_KNN + NROW*4)          // +1 MiB
#define OFF_XPRE  (OFF_SUMS + 1024)           // 256-B aligned; 134 MiB of f32

// ---------------------------------------------------------------------------
// 1) Farthest point sampling: one block per batch. 8 points/thread in regs.
//    Argmax: wave32 shuffle reduce -> 32 partials in LDS -> wave0 reduce.
// ---------------------------------------------------------------------------
__global__ __launch_bounds__(1024) void fps_kernel(const float* __restrict__ points,
                                                   int* __restrict__ p2_idx,
                                                   float* __restrict__ out_p2) {
    const int b   = blockIdx.x;
    const int tid = threadIdx.x;
    const float* pb = points + (size_t)b * N * 3;

    float px[8], py[8], pz[8], dist[8];
#pragma unroll
    for (int j = 0; j < 8; ++j) {
        int p = tid + j * 1024;
        px[j] = pb[p*3+0]; py[j] = pb[p*3+1]; pz[j] = pb[p*3+2];
        dist[j] = 1e10f;
    }

    __shared__ float red_d[32];
    __shared__ int   red_i[32];
    __shared__ int   s_cur;

    int cur = 0;
    for (int i = 0; i < NP; ++i) {
        float cx = pb[cur*3+0], cy = pb[cur*3+1], cz = pb[cur*3+2];
        if (tid == 0) {
            p2_idx[b*NP + i] = cur;
            out_p2[((size_t)b*NP + i)*3 + 0] = cx;
            out_p2[((size_t)b*NP + i)*3 + 1] = cy;
            out_p2[((size_t)b*NP + i)*3 + 2] = cz;
        }
        float bd = -1.0f; int bidx = 0x7fffffff;
#pragma unroll
        for (int j = 0; j < 8; ++j) {
            int p = tid + j * 1024;
            float dx = px[j]-cx, dy = py[j]-cy, dz = pz[j]-cz;
            float d = dx*dx + dy*dy + dz*dz;
            d = fminf(d, dist[j]);
            dist[j] = d;
            if (d > bd || (d == bd && p < bidx)) { bd = d; bidx = p; }
        }
        // wave32 shuffle argmax (first-index tie-break like jnp.argmax)
#pragma unroll
        for (int off = 16; off >= 1; off >>= 1) {
            float od = __shfl_down(bd, (unsigned)off, 32);
            int   oi = __shfl_down(bidx, (unsigned)off, 32);
            if (od > bd || (od == bd && oi < bidx)) { bd = od; bidx = oi; }
        }
        if ((tid & 31) == 0) { red_d[tid >> 5] = bd; red_i[tid >> 5] = bidx; }
        __syncthreads();
        if (tid < 32) {
            bd = red_d[tid]; bidx = red_i[tid];
#pragma unroll
            for (int off = 16; off >= 1; off >>= 1) {
                float od = __shfl_down(bd, (unsigned)off, 32);
                int   oi = __shfl_down(bidx, (unsigned)off, 32);
                if (od > bd || (od == bd && oi < bidx)) { bd = od; bidx = oi; }
            }
            if (tid == 0) s_cur = bidx;
        }
        __syncthreads();
        cur = s_cur;
    }
}

// ---------------------------------------------------------------------------
// 2) Brute-force kNN (k=16): one thread per query, points tiled through LDS,
//    sorted insertion list in registers (stable for ties like lax.top_k).
// ---------------------------------------------------------------------------
__global__ __launch_bounds__(256) void knn_kernel(const float* __restrict__ points,
                                                  const float* __restrict__ p2,
                                                  int* __restrict__ knn_idx) {
    const int qg = blockIdx.x * 256 + threadIdx.x;  // 0 .. B*NP-1 (blocks never straddle batches)
    const int b  = qg >> 11;
    const float* pb = points + (size_t)b * N * 3;
    const float qx = p2[qg*3+0], qy = p2[qg*3+1], qz = p2[qg*3+2];

    float bd[KNNK]; int bi[KNNK];
#pragma unroll
    for (int j = 0; j < KNNK; ++j) { bd[j] = 1e30f; bi[j] = 0; }

    __shared__ float tp[256 * 3];
    for (int t = 0; t < N / 256; ++t) {
        const int base = t * 256;
        __syncthreads();
        const int p = base + threadIdx.x;
        tp[threadIdx.x*3+0] = pb[p*3+0];
        tp[threadIdx.x*3+1] = pb[p*3+1];
        tp[threadIdx.x*3+2] = pb[p*3+2];
        __syncthreads();
        for (int j = 0; j < 256; ++j) {
            float dx = tp[j*3+0]-qx, dy = tp[j*3+1]-qy, dz = tp[j*3+2]-qz;
            float d = dx*dx + dy*dy + dz*dz;
            if (d < bd[KNNK-1]) {
                int s = KNNK - 1;
                while (s > 0 && bd[s-1] > d) { bd[s] = bd[s-1]; bi[s] = bi[s-1]; --s; }
                bd[s] = d; bi[s] = base + j;
            }
        }
    }
#pragma unroll
    for (int j = 0; j < KNNK; ++j) knn_idx[(size_t)qg * KNNK + j] = bi[j];
}

// ---------------------------------------------------------------------------
// 3) GEMM: x_pre = concat(points[knn], feats[knn]) @ W + b_fc.
//    One wave computes TPW consecutive 16x128 tiles with V_WMMA_F32_16X16X4_F32.
//    LDS: transposed padded W (128x68) staged once per block + per-wave A tile.
//    All A/B fragments are single aligned ds_load_b64 (bank-conflict-free).
// ---------------------------------------------------------------------------
__global__ __launch_bounds__(128) void gemm_kernel(const float* __restrict__ points,
                                                   const float* __restrict__ feats,
                                                   const float* __restrict__ Wm,
                                                   const float* __restrict__ bfc,
                                                   const int* __restrict__ knn_idx,
                                                   float* __restrict__ xpre) {
    __shared__ float sWt[COUT * KPAD];     // transposed W: sWt[n*KPAD + k]; 34816 B
    __shared__ float sA[4][16 * KPAD];     // per-wave A tile; 17408 B
    __shared__ int   sIdx[4][16];

    const int tid  = threadIdx.x;
    const int wave = tid >> 5;
    const int lane = tid & 31;

    // stage transposed W (coalesced global reads, scattered LDS writes)
    for (int i = tid; i < KDIM * COUT; i += 128) {
        int k = i >> 7, n = i & 127;
        sWt[n * KPAD + k] = Wm[i];
    }
    sWt[tid * KPAD + KDIM] = 0.0f;         // zero the K padding (tid == n)

    const int half = lane >> 4;            // K-pair select for A/B frags
    const int mr   = lane & 15;            // M for A frag, N-within-tile for B/C/D
    float* sAw = &sA[wave][0];

    for (int t = 0; t < TPW; ++t) {
        const int tile = (blockIdx.x * 4 + wave) * TPW + t;  // == b*NP + q
        const int bq   = tile >> 11;                         // batch
        const int r0   = tile * 16;                          // first GEMM row
        if (lane < 16) sIdx[wave][lane] = knn_idx[r0 + lane];
        __syncthreads();

        const float* pb = points + (size_t)bq * N * 3;
        const float* fb = feats  + (size_t)bq * N * CIN;
        for (int i = lane; i < 16 * KPAD; i += 32) {  // 34 uniform iterations
            int m = i / KPAD, c = i - m * KPAD;
            int g = sIdx[wave][m];
            float v;
            if (c < 3)          v = pb[g*3 + c];
            else if (c < KDIM)  v = fb[(size_t)g * CIN + (c - 3)];
            else                v = 0.0f;
            sAw[i] = v;
        }
        __syncthreads();

        v8f acc[8] = {};
        for (int s = 0; s < KPAD / 4; ++s) {
            const int kb = s * 4 + 2 * half;              // even -> 8B-aligned b64
            const v2f a = *reinterpret_cast<const v2f*>(sAw + mr * KPAD + kb);
#pragma unroll
            for (int nt = 0; nt < 8; ++nt) {
                const v2f bf = *reinterpret_cast<const v2f*>(
                    sWt + (nt * 16 + mr) * KPAD + kb);
                acc[nt] = __builtin_amdgcn_wmma_f32_16x16x4_f32(
                    false, a, false, bf, (short)0, acc[nt], false, false);
            }
        }

        // epilogue: +bias, store pre-BN activations
#pragma unroll
        for (int nt = 0; nt < 8; ++nt) {
            const int n = nt * 16 + mr;
            const float bias = bfc[n];
#pragma unroll
            for (int v = 0; v < 8; ++v) {
                const int m = v + 8 * half;               // C/D layout: M = vgpr + 8*half
                xpre[(size_t)(r0 + m) * COUT + n] = acc[nt][v] + bias;
            }
        }
        __syncthreads();   // protect sIdx/sA reuse next iteration
    }
}

// ---------------------------------------------------------------------------
// 4) BN stats: stride is a multiple of COUT -> each thread owns one channel;
//    private accumulate, one LDS atomic per thread, one global atomic/channel.
// ---------------------------------------------------------------------------
__global__ __launch_bounds__(256) void reduce_kernel(const float* __restrict__ xpre,
                                                     float* __restrict__ sums) {
    __shared__ float ss[COUT], sq[COUT];
    const int tid = threadIdx.x;
    if (tid < COUT) { ss[tid] = 0.0f; sq[tid] = 0.0f; }
    __syncthreads();

    const size_t total  = (size_t)NROW * COUT;
    const size_t stride = (size_t)gridDim.x * 256;   // multiple of COUT
    float ls = 0.0f, lq = 0.0f;
    for (size_t i = (size_t)blockIdx.x * 256 + tid; i < total; i += stride) {
        float v = xpre[i];
        ls += v; lq += v * v;
    }
    const int c = tid & (COUT - 1);
    atomicAdd(&ss[c], ls);
    atomicAdd(&sq[c], lq);
    __syncthreads();
    if (tid < COUT) {
        atomicAdd(&sums[tid], ss[tid]);
        atomicAdd(&sums[COUT + tid], sq[tid]);
    }
}

// ---------------------------------------------------------------------------
// 5) BN (training stats) + exact GELU + max over k=16.
// ---------------------------------------------------------------------------
__global__ __launch_bounds__(256) void finalize_kernel(const float* __restrict__ xpre,
                                                       const float* __restrict__ sums,
                                                       const float* __restrict__ gamma,
                                                       const float* __restrict__ beta,
                                                       float* __restrict__ out) {
    const int t  = blockIdx.x * 256 + threadIdx.x;   // 0 .. B*NP*COUT-1
    const int c  = t & (COUT - 1);
    const int qg = t >> 7;                           // 0 .. B*NP-1
    const float cnt  = (float)NROW;
    const float mean = sums[c] / cnt;
    const float var  = sums[COUT + c] / cnt - mean * mean;
    const float rs   = rsqrtf(var + 1e-5f);
    const float g    = gamma[c], be = beta[c];

    float mx = -1e30f;
#pragma unroll
    for (int kk = 0; kk < KNNK; ++kk) {
        float v = xpre[((size_t)qg * KNNK + kk) * COUT + c];
        float y = (v - mean) * rs * g + be;
        y = 0.5f * y * (1.0f + erff(y * 0.70710678118654752f));  // exact GELU
        mx = fmaxf(mx, y);
    }
    out[t] = mx;
}

// ---------------------------------------------------------------------------
extern "C" void kernel_launch(void* const* d_in, const int* in_sizes, int n_in,
                              void* d_out, int out_size, void* d_ws, size_t ws_size,
                              hipStream_t stream) {
    const float* feats  = (const float*)d_in[0];
    const float* points = (const float*)d_in[1];
    const float* Wm     = (const float*)d_in[2];
    const float* bfc    = (const float*)d_in[3];
    const float* gamma  = (const float*)d_in[4];
    const float* beta   = (const float*)d_in[5];

    float* out_x  = (float*)d_out;                       // [8,2048,128]
    float* out_p2 = out_x + (size_t)B * NP * COUT;       // [8,2048,3]

    char* ws = (char*)d_ws;
    int*   p2_idx = (int*)  (ws + OFF_P2IDX);
    int*   knn    = (int*)  (ws + OFF_KNN);
    float* sums   = (float*)(ws + OFF_SUMS);
    float* xpre   = (float*)(ws + OFF_XPRE);

    hipMemsetAsync(sums, 0, 2 * COUT * sizeof(float), stream);
    fps_kernel<<<B, 1024, 0, stream>>>(points, p2_idx, out_p2);
    knn_kernel<<<(B * NP) / 256, 256, 0, stream>>>(points, out_p2, knn);
    gemm_kernel<<<NTILE / (4 * TPW), 128, 0, stream>>>(points, feats, Wm, bfc, knn, xpre);
    reduce_kernel<<<1024, 256, 0, stream>>>(xpre, sums);
    finalize_kernel<<<(B * NP * COUT) / 256, 256, 0, stream>>>(xpre, sums, gamma, beta, out_x);
}